// CenterLoss_36953898615125
// MI455X (gfx1250) — compile-verified
//
#include <hip/hip_runtime.h>

typedef float v2f __attribute__((ext_vector_type(2)));
typedef float v8f __attribute__((ext_vector_type(8)));

#define FEAT        128
#define BATCH       1024
#define NUM_CLASSES 100000

// Each wave (32 lanes) handles 16 samples. Lane L: row = tile*16 + (L&15),
// half = L>>4 selects K%4 in {0,1} (half=0) or {2,3} (half=1).
// With that layout, the 16x4 A-matrix of diffs and the 4x16 B-matrix of
// transposed diffs are register-identical, so wmma(d,d,acc) accumulates the
// Gram tile D*D^T; its diagonal is the per-sample squared distance.
__global__ __launch_bounds__(256) void center_loss_partial(
    const float* __restrict__ x,
    const int*   __restrict__ labels,
    const float* __restrict__ centers,
    float*       __restrict__ partial)
{
    const int tid  = threadIdx.x;
    const int lane = tid & 31;
    const int wave = tid >> 5;                  // 0..7
    const int tile = blockIdx.x * 8 + wave;     // 0..63, 16 samples each
    const int row  = tile * 16 + (lane & 15);   // 0..1023
    const int half = lane >> 4;                 // 0 or 1

    const float* xr = x + (size_t)row * FEAT;
    const float* cr = centers + (size_t)labels[row] * FEAT;

    v8f acc0 = {0.f, 0.f, 0.f, 0.f, 0.f, 0.f, 0.f, 0.f};
    v8f acc1 = acc0;

    #pragma unroll
    for (int t = 0; t < 32; t += 2) {
        const int k0 = t * 4 + half * 2;        // even -> 8B aligned
        const int k1 = k0 + 4;
        v2f d0 = *(const v2f*)(xr + k0) - *(const v2f*)(cr + k0);
        v2f d1 = *(const v2f*)(xr + k1) - *(const v2f*)(cr + k1);
        acc0 = __builtin_amdgcn_wmma_f32_16x16x4_f32(
            false, d0, false, d0, (short)0, acc0, false, false);
        acc1 = __builtin_amdgcn_wmma_f32_16x16x4_f32(
            false, d1, false, d1, (short)0, acc1, false, false);
    }

    // Extract Gram-tile diagonal: sample m=j at (VGPR j, lane j),
    // sample m=j+8 at (VGPR j, lane j+24).
    float v = 0.f;
    #pragma unroll
    for (int j = 0; j < 8; ++j) {
        const float dj = acc0[j] + acc1[j];
        if (lane == j)      v = dj;
        if (lane == j + 24) v = dj;
    }

    // wave32 reduction
    #pragma unroll
    for (int off = 16; off > 0; off >>= 1)
        v += __shfl_down(v, off, 32);

    __shared__ float smem[8];
    if (lane == 0) smem[wave] = v;
    __syncthreads();

    if (tid == 0) {
        float s = 0.f;
        #pragma unroll
        for (int w = 0; w < 8; ++w) s += smem[w];
        partial[blockIdx.x] = s;
    }
}

__global__ void center_loss_finalize(const float* __restrict__ partial,
                                     float* __restrict__ out)
{
    float s = 0.f;
    #pragma unroll
    for (int b = 0; b < 8; ++b) s += partial[b];  // fixed order: deterministic
    out[0] = s * (1.0f / ((float)BATCH * (float)NUM_CLASSES));
}

extern "C" void kernel_launch(void* const* d_in, const int* in_sizes, int n_in,
                              void* d_out, int out_size, void* d_ws, size_t ws_size,
                              hipStream_t stream)
{
    const float* x       = (const float*)d_in[0];  // (1024, 128) f32
    const int*   labels  = (const int*)  d_in[1];  // (1024,)     i32
    const float* centers = (const float*)d_in[2];  // (100000,128) f32
    float* partial = (float*)d_ws;                 // 8 floats of scratch
    float* out     = (float*)d_out;                // scalar f32

    center_loss_partial<<<dim3(8), dim3(256), 0, stream>>>(x, labels, centers, partial);
    center_loss_finalize<<<dim3(1), dim3(1), 0, stream>>>(partial, out);
}